// bSpline_57784490000610
// MI455X (gfx1250) — compile-verified
//
#include <hip/hip_runtime.h>

typedef float v2f __attribute__((ext_vector_type(2)));
typedef float v8f __attribute__((ext_vector_type(8)));
typedef int   b64v __attribute__((vector_size(8)));   // payload type for async b64

#define GS      1000
#define START_F (-3.0f)
#define SCALE_F 166.5f   /* (GS-1)/(END-START) = 999/6, exact in fp32 */

// Opportunistic CDNA5 async global->LDS staging (compile-safe: only used if
// the toolchain declares the builtin; falls back to plain loads otherwise).
#if defined(__AMDGCN__) && __has_builtin(__builtin_amdgcn_global_load_async_to_lds_b64)
#define HAVE_ASYNC_LDS 1
#else
#define HAVE_ASYNC_LDS 0
#endif

// ---------------------------------------------------------------------------
// Kernel 1: g[i] = dot(basis_grid[i,:], control_points)  for i in [0,1000)
// computed with V_WMMA_F32_16X16X4_F32 (exact fp32 matmul), then packed as
// (g[i], g[i+1]) pairs into global scratch so the hot loop does one b64 load.
// ---------------------------------------------------------------------------
__global__ void __launch_bounds__(256)
bspline_build_table(const float* __restrict__ cp,
                    const float* __restrict__ basis,
                    float2* __restrict__ tab)
{
    __shared__ float gtab[1024];          // rows 1000..1023 = scratch pad

    const int tid  = threadIdx.x;
    const int wave = tid >> 5;
    const int lane = tid & 31;
    const int m    = lane & 15;      // row within 16-row tile
    const int half = lane >> 4;      // 0: lanes 0-15 (K 0,1), 1: lanes 16-31 (K 2,3)

    // 63 tiles of 16 rows cover rows 0..1007; rows >= 1000 are clamped for
    // addressing; their (garbage) outputs land in the gtab pad region and are
    // never read — row results are independent in D = A x B.
    for (int t = wave; t < 63; t += 8) {
        const int row  = t * 16 + m;
        const int rowc = row < GS ? row : GS - 1;

        v8f acc = {};
#pragma unroll
        for (int kc = 0; kc < 4; ++kc) {
            const int kb = kc * 4 + 2 * half;
            v2f a, b;
            // A: 16x4 fp32 tile of basis_grid (lane-half split over K)
            a.x = basis[rowc * 16 + kb];
            a.y = basis[rowc * 16 + kb + 1];
            // B: 4x16 with cp chunk broadcast across all 16 columns
            b.x = cp[kb];
            b.y = cp[kb + 1];
            // 8 args: (neg_a, A, neg_b, B, c_mod, C, reuse_a, reuse_b)
            acc = __builtin_amdgcn_wmma_f32_16x16x4_f32(
                      false, a, false, b, (short)0, acc, false, false);
        }

        // Column N=0 of D lives in lanes 0 (rows t*16+0..7, VGPR 0..7) and
        // 16 (rows t*16+8..15). Max r = 1007 < 1024: store branchlessly.
        if (m == 0) {
#pragma unroll
            for (int j = 0; j < 8; ++j)
                gtab[t * 16 + half * 8 + j] = acc[j];
        }
    }
    __syncthreads();

    for (int i = tid; i < 1024; i += 256) {
        const float a = gtab[i     < GS ? i     : GS - 1];
        const float b = gtab[i + 1 < GS ? i + 1 : GS - 1];
        tab[i] = make_float2(a, b);
    }
}

// ---------------------------------------------------------------------------
// Kernel 2: streaming gather. out[i] = g[f] + w*(g[f+1]-g[f]).
// ---------------------------------------------------------------------------
__device__ __forceinline__ float bspline_one(float xv, const float2* gp)
{
    float xc = fminf(fmaxf(xv, -3.0f), 3.0f);
    float t  = (xc - START_F) * SCALE_F;        // t in [0, 999]
    int   f  = (int)t;                          // trunc == floor (t >= 0)
    f = f > GS - 1 ? GS - 1 : f;
    const float  w = t - (float)f;
    const float2 p = gp[f];                     // single ds_load_b64
    return fmaf(w, p.y - p.x, p.x);
}

__global__ void __launch_bounds__(256)
bspline_eval(const float4* __restrict__ x4, const float2* __restrict__ tab,
             float4* __restrict__ out4, int n4,
             const float* __restrict__ x, float* __restrict__ out, int n)
{
    __shared__ float2 gp[1024];   // 8 KB pair table

#if HAVE_ASYNC_LDS
    {
        // global_load_async_to_lds_b64: (AS1 b64v* src, AS3 b64v* dst, imm, cpol)
        auto src = (__attribute__((address_space(1))) b64v*)tab;
        auto dst = (__attribute__((address_space(3))) b64v*)gp;
        for (int i = threadIdx.x; i < 1024; i += 256)
            __builtin_amdgcn_global_load_async_to_lds_b64(src + i, dst + i, 0, 0);
#if __has_builtin(__builtin_amdgcn_s_wait_asynccnt)
        __builtin_amdgcn_s_wait_asynccnt(0);
#else
        asm volatile("s_wait_asynccnt 0x0" ::: "memory");
#endif
    }
#else
    for (int i = threadIdx.x; i < 1024; i += 256) gp[i] = tab[i];
#endif
    __syncthreads();

    const int stride = gridDim.x * blockDim.x;
    for (int i = blockIdx.x * blockDim.x + threadIdx.x; i < n4; i += stride) {
        __builtin_prefetch(&x4[i + stride], 0, 0);   // global_prefetch_b8
        const float4 v = x4[i];
        float4 r;
        r.x = bspline_one(v.x, gp);
        r.y = bspline_one(v.y, gp);
        r.z = bspline_one(v.z, gp);
        r.w = bspline_one(v.w, gp);
        out4[i] = r;
    }

    // scalar tail (empty for n = 2^24, kept for generality)
    const int tail = n - (n4 << 2);
    if (blockIdx.x == 0 && (int)threadIdx.x < tail) {
        const int idx = (n4 << 2) + threadIdx.x;
        out[idx] = bspline_one(x[idx], gp);
    }
}

// ---------------------------------------------------------------------------
extern "C" void kernel_launch(void* const* d_in, const int* in_sizes, int n_in,
                              void* d_out, int out_size, void* d_ws, size_t ws_size,
                              hipStream_t stream)
{
    const float* x     = (const float*)d_in[0];
    const float* cp    = (const float*)d_in[1];
    const float* basis = (const float*)d_in[2];
    float*  out = (float*)d_out;
    float2* tab = (float2*)d_ws;            // needs 8 KB of scratch
    const int n = in_sizes[0];

    bspline_build_table<<<1, 256, 0, stream>>>(cp, basis, tab);

    const int n4 = n >> 2;
    int blocks = (n4 + 255) / 256;
    if (blocks > 2048) blocks = 2048;
    if (blocks < 1)    blocks = 1;
    bspline_eval<<<blocks, 256, 0, stream>>>((const float4*)x, tab,
                                             (float4*)out, n4, x, out, n);
}